// Generator_69320772158392
// MI455X (gfx1250) — compile-verified
//
#include <hip/hip_runtime.h>
#include <math.h>

typedef float v2f __attribute__((ext_vector_type(2)));
typedef float v8f __attribute__((ext_vector_type(8)));
typedef unsigned v4u __attribute__((ext_vector_type(4)));
typedef int v4i __attribute__((ext_vector_type(4)));
typedef int v8i __attribute__((ext_vector_type(8)));

#define LRELU_S 0.2f
#define EPSV 1e-8f
#define KTILE 288  // 32*9; divides IC*9 for all IC in {64,128,256,512,1024}

#if __has_builtin(__builtin_amdgcn_tensor_load_to_lds)
#define USE_TDM 1
#else
#define USE_TDM 0
#endif

// ---------------------------------------------------------------------------
// Full-precision matrix core: V_WMMA_F32_16X16X4_F32 (D = A(16x4) x B(4x16) + C)
// ---------------------------------------------------------------------------
__device__ __forceinline__ v8f wmma_f32_k4(v2f a, v2f b, v8f c) {
  return __builtin_amdgcn_wmma_f32_16x16x4_f32(false, a, false, b, (short)0, c,
                                               false, false);
}

__device__ __forceinline__ float lrelu_f(float x) { return x >= 0.f ? x : LRELU_S * x; }

// Deterministic hash-based gaussian (stand-in for jax.random.normal noise)
__device__ __forceinline__ float gnoise(unsigned seed, unsigned idx) {
  unsigned s = (seed ^ 0xA511E9B3u) * 0x9E3779B9u + idx * 0x85EBCA6Bu + 0x165667B1u;
  s ^= s >> 16; s *= 0x7FEB352Du; s ^= s >> 15; s *= 0x846CA68Bu; s ^= s >> 16;
  unsigned t = (s ^ idx) * 0x2C1B3C6Du + 0x297A2D39u;
  t ^= t >> 15; t *= 0x9E3779B9u; t ^= t >> 16;
  float u1 = ((s >> 8) + 1u) * (1.f / 16777216.f);
  float u2 = (t >> 8) * (1.f / 16777216.f);
  return sqrtf(-2.f * __logf(u1)) * __cosf(6.28318530718f * u2);
}

#if USE_TDM
// ---------------------------------------------------------------------------
// TDM: DMA a 2D tile (16 rows x KTILE f32, row stride ktot) from global to LDS.
// D# bitfields per CDNA5 ISA ch.8 (group0 128b, group1 256b; groups 2/3 unused
// for 2D tensors). All inputs are wave-uniform. 6-arg clang-23 builtin form.
// ---------------------------------------------------------------------------
__device__ __forceinline__ void tdm_load_wtile(const float* gaddr, unsigned lds_off,
                                               int ktot, int oc_total) {
  unsigned long long ga = (unsigned long long)(size_t)gaddr;
  v4u g0;
  g0.x = 1u;                                                // count=1, user desc
  g0.y = lds_off;                                           // lds_addr[31:0]
  g0.z = (unsigned)(ga & 0xFFFFFFFFu);                      // global_addr[31:0]
  g0.w = (unsigned)((ga >> 32) & 0x01FFFFFFu) | (2u << 30); // ga[56:32] | type=2
  unsigned td0 = (unsigned)ktot;      // tensor dim0 (elements per row)
  unsigned td1 = (unsigned)oc_total;  // tensor dim1 (rows)
  v8i g1;
  g1[0] = (int)(2u << 16);                                   // data_size=4B
  g1[1] = (int)((td0 & 0xFFFFu) << 16);                      // tensor_dim0[15:0]
  g1[2] = (int)((td0 >> 16) | ((td1 & 0xFFFFu) << 16));      // dim0[31:16]|dim1[15:0]
  g1[3] = (int)((td1 >> 16) | ((unsigned)KTILE << 16));      // dim1[31:16]|tile_dim0
  g1[4] = 16;                                                // tile_dim1=16, tile_dim2=0
  g1[5] = (int)td0;                                          // tensor_dim0_stride[31:0]
  g1[6] = 0;
  g1[7] = 0;
  v4i zz4 = {0, 0, 0, 0};
  v8i zz8 = {0, 0, 0, 0, 0, 0, 0, 0};
  __builtin_amdgcn_tensor_load_to_lds(g0, g1, zz4, zz4, zz8, 0);
}
#endif

// ---------------------------------------------------------------------------
// Mapping MLP: 8 x [Linear + LeakyReLU] + pixel-norm. One WG, 8 waves.
// Batch padded 4->16; pad rows carry don't-care data (WMMA rows independent).
// ---------------------------------------------------------------------------
struct MapArgs { const float* W[8]; const float* b[8]; };

__global__ __launch_bounds__(256) void mapping_kernel(const float* __restrict__ z,
                                                      MapArgs args,
                                                      float* __restrict__ wlat) {
  __shared__ float xb[2][16 * 512];
  __shared__ float rs[16];
  int t = threadIdx.x;
  int lane = t & 31, wv = t >> 5;
  int m = lane & 15, hi = lane >> 4;
  for (int i = t; i < 16 * 512; i += 256) xb[0][i] = 0.f;
  __syncthreads();
  for (int i = t; i < 4 * 128; i += 256) xb[0][(i >> 7) * 512 + (i & 127)] = z[i];
  __syncthreads();
  int cur = 0;
  for (int layer = 0; layer < 8; ++layer) {
    int K = (layer == 0) ? 128 : 512;
    const float* W = args.W[layer];
    const float* bias = args.b[layer];
    for (int nt = wv; nt < 32; nt += 8) {
      int n0 = nt * 16;
      v8f acc = {};
      #pragma unroll 4
      for (int k0 = 0; k0 < K; k0 += 4) {
        int kk = k0 + 2 * hi;
        v2f a;  a.x = xb[cur][m * 512 + kk];     a.y = xb[cur][m * 512 + kk + 1];
        v2f bf; bf.x = W[(n0 + m) * K + kk];     bf.y = W[(n0 + m) * K + kk + 1];
        acc = wmma_f32_k4(a, bf, acc);
      }
      float bv = bias[n0 + m];
      #pragma unroll
      for (int v = 0; v < 8; ++v)
        xb[cur ^ 1][(v + 8 * hi) * 512 + n0 + m] = lrelu_f(acc[v] + bv);
    }
    __syncthreads();
    cur ^= 1;
  }
  if (t < 16) {
    float s = 0.f;
    for (int c = 0; c < 512; ++c) { float v = xb[cur][t * 512 + c]; s += v * v; }
    rs[t] = rsqrtf(s * (1.f / 512.f) + EPSV);
  }
  __syncthreads();
  for (int i = t; i < 16 * 512; i += 256) wlat[i] = xb[cur][i] * rs[i >> 9];
}

// ---------------------------------------------------------------------------
// styles[b, n] = wlat[b, :] . aw[n, :] + ab[n]   (16 rows; rows >=4 don't-care)
// ---------------------------------------------------------------------------
__global__ __launch_bounds__(32) void style_kernel(const float* __restrict__ wlat,
                                                   const float* __restrict__ aw,
                                                   const float* __restrict__ ab,
                                                   float* __restrict__ styles, int C) {
  int lane = threadIdx.x, m = lane & 15, hi = lane >> 4;
  int n0 = blockIdx.x * 16;
  v8f acc = {};
  #pragma unroll 4
  for (int k0 = 0; k0 < 512; k0 += 4) {
    int kk = k0 + 2 * hi;
    v2f a;  a.x = wlat[m * 512 + kk];          a.y = wlat[m * 512 + kk + 1];
    v2f bf; bf.x = aw[(n0 + m) * 512 + kk];    bf.y = aw[(n0 + m) * 512 + kk + 1];
    acc = wmma_f32_k4(a, bf, acc);
  }
  float bv = ab[n0 + m];
  #pragma unroll
  for (int v = 0; v < 8; ++v)
    styles[(v + 8 * hi) * C + n0 + m] = acc[v] + bv;
}

// demod[b,o] = rsqrt( sum_i s[b,i]^2 * sum_kk W[o,i,kk]^2 + eps )
__global__ void demod_kernel(const float* __restrict__ W, const float* __restrict__ styles,
                             float* __restrict__ demod, int IC, int OC) {
  int i = blockIdx.x * blockDim.x + threadIdx.x;
  if (i >= 4 * OC) return;
  int b = i / OC, o = i - b * OC;
  float s = 0.f;
  for (int c = 0; c < IC; ++c) {
    const float* wp = W + ((long)o * IC + c) * 9;
    float wsq = 0.f;
    #pragma unroll
    for (int k = 0; k < 9; ++k) wsq += wp[k] * wp[k];
    float st = styles[b * IC + c];
    s += st * st * wsq;
  }
  demod[b * OC + o] = rsqrtf(s + EPSV);
}

// nearest 2x upsample (bstride==0 broadcasts the learned const over batch)
__global__ void up2_kernel(const float* __restrict__ in, float* __restrict__ out,
                           int C, int Hin, int bstride) {
  int Ho = Hin * 2;
  int total = 4 * C * Ho * Ho;
  int i = blockIdx.x * 256 + threadIdx.x;
  if (i >= total) return;
  int x = i % Ho;
  int y = (i / Ho) % Ho;
  int c = (i / (Ho * Ho)) % C;
  int b = i / (Ho * Ho * C);
  out[i] = in[(((long)b * bstride + c) * Hin + (y >> 1)) * Hin + (x >> 1)];
}

__device__ __forceinline__ float conv_b_elem(const float* __restrict__ xb,
                                             const float* __restrict__ sb,
                                             int K, int py, int px, int H, int Wd, int HW) {
  int ic = K / 9;
  int r = K - ic * 9;
  int yy = py + r / 3 - 1;
  int xx = px + (r - (r / 3) * 3) - 1;
  if ((unsigned)yy < (unsigned)H && (unsigned)xx < (unsigned)Wd)
    return xb[(long)ic * HW + yy * Wd + xx] * sb[ic];
  return 0.f;
}

// ---------------------------------------------------------------------------
// Modulated 3x3 conv as implicit GEMM: M=oc tile, N=16 pixels, K=ic*9.
// Weights streamed by TDM (tensor_load_to_lds) double-buffered in LDS;
// activations gathered (style-modulated, zero-padded) as B fragments.
// Epilogue fuses demod + noise + LeakyReLU. One wave per 16x16 output tile.
// ---------------------------------------------------------------------------
__global__ __launch_bounds__(32) void modconv3x3_kernel(
    const float* __restrict__ x, const float* __restrict__ W,
    const float* __restrict__ styles, const float* __restrict__ demod,
    const float* __restrict__ nscale, float* __restrict__ out,
    int IC, int OC, int H, int Wd, unsigned seed) {
#if USE_TDM
  __shared__ float wtile[2][16 * KTILE];
#endif
  int lane = threadIdx.x, m = lane & 15, hi = lane >> 4;
  int HW = H * Wd;
  int n0 = blockIdx.x * 16;
  int o0 = blockIdx.y * 16;
  int b = n0 / HW;                 // HW multiple of 16 -> tiles never cross batch
  int p = (n0 - b * HW) + m;
  int py = p / Wd, px = p - py * Wd;
  int Ktot = IC * 9;
  const float* xb = x + (long)b * IC * HW;
  const float* sb = styles + b * IC;
  const float* Wr = W + (long)(o0 + m) * Ktot;
  int nchunks = Ktot / KTILE;
  v8f acc = {};
#if USE_TDM
  unsigned lds0 = (unsigned)(size_t)(void*)&wtile[0][0];
  unsigned lds1 = (unsigned)(size_t)(void*)&wtile[1][0];
  tdm_load_wtile(W + (size_t)o0 * Ktot, lds0, Ktot, OC);
#endif
  for (int ch = 0; ch < nchunks; ++ch) {
    int kbase = ch * KTILE;
#if USE_TDM
    if (ch + 1 < nchunks) {
      tdm_load_wtile(W + (size_t)o0 * Ktot + (kbase + KTILE),
                     ((ch & 1) ? lds0 : lds1), Ktot, OC);
      __builtin_amdgcn_s_wait_tensorcnt(1);   // current chunk's DMA done
    } else {
      __builtin_amdgcn_s_wait_tensorcnt(0);
    }
    const float* lw = &wtile[ch & 1][0];
#endif
    #pragma unroll 2
    for (int k0 = 0; k0 < KTILE; k0 += 4) {
      int kk = k0 + 2 * hi;
      v2f a;
#if USE_TDM
      a.x = lw[m * KTILE + kk];
      a.y = lw[m * KTILE + kk + 1];
#else
      a.x = Wr[kbase + kk];
      a.y = Wr[kbase + kk + 1];
      __builtin_prefetch(Wr + kbase + kk + 64, 0, 0);
#endif
      int K = kbase + kk;
      v2f bf; bf.x = conv_b_elem(xb, sb, K,     py, px, H, Wd, HW);
              bf.y = conv_b_elem(xb, sb, K + 1, py, px, H, Wd, HW);
      acc = wmma_f32_k4(a, bf, acc);
    }
  }
  (void)Wr;
  #pragma unroll
  for (int v = 0; v < 8; ++v) {
    int oc = o0 + v + 8 * hi;
    float y = acc[v] * demod[b * OC + oc];
    unsigned idx = (unsigned)(b * OC + oc) * (unsigned)HW + (unsigned)p;
    y += nscale[oc] * gnoise(seed, idx);
    out[((long)(b * OC + oc)) * HW + p] = lrelu_f(y);
  }
}

// toRGB: 1x1 modulated conv, OC=1, no demod/noise/act
__global__ void torgb_kernel(const float* __restrict__ x, const float* __restrict__ w,
                             const float* __restrict__ styles, float* __restrict__ rgb,
                             int C, int HW) {
  int i = blockIdx.x * 256 + threadIdx.x;
  if (i >= 4 * HW) return;
  int b = i / HW, p = i - b * HW;
  float acc = 0.f;
  for (int c = 0; c < C; ++c)
    acc += w[c] * x[((long)(b * C + c)) * HW + p] * styles[b * C + c];
  rgb[i] = acc;
}

__global__ void rgb_combine_kernel(const float* __restrict__ prev,
                                   const float* __restrict__ curr,
                                   float* __restrict__ out, int H, int first) {
  int total = 4 * H * H;
  int i = blockIdx.x * 256 + threadIdx.x;
  if (i >= total) return;
  if (first) { out[i] = curr[i]; return; }
  int x = i % H, y = (i / H) % H, b = i / (H * H);
  int Hp = H >> 1;
  float pv = prev[(b * Hp + (y >> 1)) * Hp + (x >> 1)];
  out[i] = (pv + curr[i]) * 0.70710678f;
}

__global__ void tanh_kernel(const float* __restrict__ rgb, float* __restrict__ out, int n) {
  int i = blockIdx.x * 256 + threadIdx.x;
  if (i >= n) return;
  float v = rgb[i];
  v = fminf(fmaxf(v, -10.f), 10.f);
  out[i] = tanhf(v);
}

// ---------------------------------------------------------------------------
// Spectral norm: sigma = || M @ normalize(M^T u) ||  (one thread per matrix)
// ---------------------------------------------------------------------------
__global__ void sigma_kernel(const float* qw, const float* kw, const float* vw, const float* ow,
                             const float* qu, const float* ku, const float* vu, const float* ou,
                             float* __restrict__ sigma) {
  int t = threadIdx.x;
  if (t >= 4) return;
  const float* w; const float* u; int R, C;
  if (t == 0)      { w = qw; u = qu; R = 8;  C = 64; }
  else if (t == 1) { w = kw; u = ku; R = 8;  C = 64; }
  else if (t == 2) { w = vw; u = vu; R = 32; C = 64; }
  else             { w = ow; u = ou; R = 64; C = 32; }
  float vvec[64];
  float vn = 0.f;
  for (int c = 0; c < C; ++c) {
    float s = 0.f;
    for (int r = 0; r < R; ++r) s += w[r * C + c] * u[r];
    vvec[c] = s; vn += s * s;
  }
  float inv = 1.f / (sqrtf(vn) + 1e-12f);
  float sn = 0.f;
  for (int r = 0; r < R; ++r) {
    float s = 0.f;
    for (int c = 0; c < C; ++c) s += w[r * C + c] * vvec[c];
    s *= inv; sn += s * s;
  }
  sigma[t] = sqrtf(sn);
}

// Fused QKV 1x1 projections with spectral-norm scaling
__global__ void qkv_kernel(const float* __restrict__ x,
                           const float* __restrict__ qw, const float* __restrict__ qb,
                           const float* __restrict__ kw, const float* __restrict__ kb,
                           const float* __restrict__ vw, const float* __restrict__ vb,
                           const float* __restrict__ sigma,
                           float* __restrict__ qbuf, float* __restrict__ kbuf,
                           float* __restrict__ vbuf) {
  int i = blockIdx.x * 256 + threadIdx.x;
  if (i >= 4 * 4096) return;
  int b = i >> 12, p = i & 4095;
  const float* xb = x + ((long)b * 64) * 4096 + p;
  float isq = 1.f / sigma[0], isk = 1.f / sigma[1], isv = 1.f / sigma[2];
  float qa[8] = {}, ka[8] = {}, va[32] = {};
  for (int c = 0; c < 64; ++c) {
    float xv = xb[(long)c * 4096];
    #pragma unroll
    for (int d = 0; d < 8; ++d) { qa[d] += qw[d * 64 + c] * xv; ka[d] += kw[d * 64 + c] * xv; }
    #pragma unroll
    for (int d = 0; d < 32; ++d) va[d] += vw[d * 64 + c] * xv;
  }
  #pragma unroll
  for (int d = 0; d < 8; ++d) {
    qbuf[(long)i * 8 + d] = qa[d] * isq + qb[d];
    kbuf[(long)i * 8 + d] = ka[d] * isk + kb[d];
  }
  #pragma unroll
  for (int d = 0; d < 32; ++d) vbuf[(long)i * 32 + d] = va[d] * isv + vb[d];
}

// ---------------------------------------------------------------------------
// Flash attention: per wave, 16 queries; stream 4096 keys in 16-tiles.
// S = Q K^T via WMMA, online softmax (16-lane shuffles), P through LDS to
// A-fragment layout, O += P V via WMMA (16x32 accumulator).
// ---------------------------------------------------------------------------
__global__ __launch_bounds__(32) void attn_flash_kernel(const float* __restrict__ qbuf,
                                                        const float* __restrict__ kbuf,
                                                        const float* __restrict__ vbuf,
                                                        float* __restrict__ obuf) {
  __shared__ float pls[16 * 16];
  int lane = threadIdx.x, m = lane & 15, hi = lane >> 4;
  int q0 = blockIdx.x * 16;
  int b = blockIdx.y;
  const float* qb_ = qbuf + ((long)b * 4096) * 8;
  const float* kb_ = kbuf + ((long)b * 4096) * 8;
  const float* vb_ = vbuf + ((long)b * 4096) * 32;
  float mrow[8], lrow[8];
  v8f o0 = {}, o1 = {};
  #pragma unroll
  for (int v = 0; v < 8; ++v) { mrow[v] = -1e30f; lrow[v] = 0.f; }
  const float scale = 0.35355339059f;  // 1/sqrt(8)
  for (int k0 = 0; k0 < 4096; k0 += 16) {
    v8f s = {};
    #pragma unroll
    for (int kq = 0; kq < 8; kq += 4) {
      int kk = kq + 2 * hi;
      v2f a;  a.x = qb_[(q0 + m) * 8 + kk];  a.y = qb_[(q0 + m) * 8 + kk + 1];
      v2f bb; bb.x = kb_[(k0 + m) * 8 + kk]; bb.y = kb_[(k0 + m) * 8 + kk + 1];
      s = wmma_f32_k4(a, bb, s);
    }
    float pv[8];
    #pragma unroll
    for (int v = 0; v < 8; ++v) {
      float sv = s[v] * scale;
      float tmax = sv;
      for (int i2 = 1; i2 < 16; i2 <<= 1) tmax = fmaxf(tmax, __shfl_xor(tmax, i2, 32));
      float nm = fmaxf(mrow[v], tmax);
      float corr = __expf(mrow[v] - nm);
      float pp = __expf(sv - nm);
      float ts = pp;
      for (int i2 = 1; i2 < 16; i2 <<= 1) ts += __shfl_xor(ts, i2, 32);
      lrow[v] = lrow[v] * corr + ts;
      mrow[v] = nm;
      o0[v] *= corr; o1[v] *= corr;
      pv[v] = pp;
    }
    #pragma unroll
    for (int v = 0; v < 8; ++v) pls[(v + 8 * hi) * 16 + m] = pv[v];
    __syncthreads();
    #pragma unroll
    for (int kq = 0; kq < 16; kq += 4) {
      int kk = kq + 2 * hi;
      v2f a;  a.x  = pls[m * 16 + kk];             a.y  = pls[m * 16 + kk + 1];
      v2f b0; b0.x = vb_[(k0 + kk) * 32 + m];      b0.y = vb_[(k0 + kk + 1) * 32 + m];
      v2f b1; b1.x = vb_[(k0 + kk) * 32 + 16 + m]; b1.y = vb_[(k0 + kk + 1) * 32 + 16 + m];
      o0 = wmma_f32_k4(a, b0, o0);
      o1 = wmma_f32_k4(a, b1, o1);
    }
    __syncthreads();
  }
  #pragma unroll
  for (int v = 0; v < 8; ++v) {
    float inv = 1.f / lrow[v];
    int p = q0 + v + 8 * hi;
    obuf[((long)(b * 4096) + p) * 32 + m]      = o0[v] * inv;
    obuf[((long)(b * 4096) + p) * 32 + 16 + m] = o1[v] * inv;
  }
}

// out = x + gamma * (o @ (o_w/sigma)^T + o_b)
__global__ void attn_out_kernel(const float* __restrict__ x, const float* __restrict__ obuf,
                                const float* __restrict__ ow, const float* __restrict__ ob,
                                const float* __restrict__ sigma, const float* __restrict__ gamma,
                                float* __restrict__ out) {
  int total = 4 * 64 * 4096;
  int i = blockIdx.x * 256 + threadIdx.x;
  if (i >= total) return;
  int p = i & 4095;
  int c = (i >> 12) & 63;
  int b = i >> 18;
  float iso = 1.f / sigma[3];
  const float* ov = obuf + ((long)(b * 4096) + p) * 32;
  float acc = 0.f;
  #pragma unroll
  for (int d = 0; d < 32; ++d) acc += ow[c * 32 + d] * ov[d];
  out[i] = x[i] + gamma[0] * (acc * iso + ob[c]);
}

// ---------------------------------------------------------------------------
// Host orchestration
// ---------------------------------------------------------------------------
extern "C" void kernel_launch(void* const* d_in, const int* in_sizes, int n_in,
                              void* d_out, int out_size, void* d_ws, size_t ws_size,
                              hipStream_t stream) {
  (void)in_sizes; (void)n_in; (void)out_size; (void)ws_size;
  const float* z = (const float*)d_in[0];
  MapArgs ma;
  for (int i = 0; i < 8; ++i) { ma.W[i] = (const float*)d_in[1 + i]; ma.b[i] = (const float*)d_in[9 + i]; }
  const float* cst = (const float*)d_in[17];

  static const int IN_CHS[5]  = {1024, 512, 256, 128, 64};
  static const int OUT_CHS[5] = {512, 256, 128, 64, 64};

  float* ws = (float*)d_ws;
  size_t off = 0;
  float* wlat   = ws + off; off += 16 * 512;
  float* styles = ws + off; off += 16 * 1024;
  float* demod  = ws + off; off += 4 * 512;
  float* sigma  = ws + off; off += 16;
  float* bufA   = ws + off; off += 4u * 64 * 128 * 128;
  float* bufB   = ws + off; off += 4u * 64 * 128 * 128;
  float* rgb0   = ws + off; off += 4 * 128 * 128;
  float* rgb1   = ws + off; off += 4 * 128 * 128;
  float* rgbT   = ws + off; off += 4 * 128 * 128;
  float* qbuf   = ws + off; off += 4 * 4096 * 8;
  float* kbuf   = ws + off; off += 4 * 4096 * 8;
  float* vbuf   = ws + off; off += 4 * 4096 * 32;
  float* obuf   = ws + off; off += 4 * 4096 * 32;

  mapping_kernel<<<1, 256, 0, stream>>>(z, ma, wlat);

  float* xin = nullptr;
  float* rgb_cur = rgb0;
  float* rgb_alt = rgb1;

  for (int i = 0; i < 5; ++i) {
    int ic = IN_CHS[i], oc = OUT_CHS[i];
    int Hin = 4 << i, H = 8 << i, HW = H * H;
    const float* c1w  = (const float*)d_in[18 + 8 * i + 0];
    const float* c1aw = (const float*)d_in[18 + 8 * i + 1];
    const float* c1ab = (const float*)d_in[18 + 8 * i + 2];
    const float* c2w  = (const float*)d_in[18 + 8 * i + 3];
    const float* c2aw = (const float*)d_in[18 + 8 * i + 4];
    const float* c2ab = (const float*)d_in[18 + 8 * i + 5];
    const float* n1   = (const float*)d_in[18 + 8 * i + 6];
    const float* n2   = (const float*)d_in[18 + 8 * i + 7];
    const float* tgw  = (const float*)d_in[58 + 3 * i + 0];
    const float* tgaw = (const float*)d_in[58 + 3 * i + 1];
    const float* tgab = (const float*)d_in[58 + 3 * i + 2];

    float* u   = (i == 0) ? bufA : ((xin == bufA) ? bufB : bufA);
    float* tmp = (u == bufA) ? bufB : bufA;

    int upTot = 4 * ic * HW;
    up2_kernel<<<(upTot + 255) / 256, 256, 0, stream>>>(
        (i == 0) ? cst : xin, u, ic, Hin, (i == 0) ? 0 : ic);

    // conv1: u -> tmp
    style_kernel<<<ic / 16, 32, 0, stream>>>(wlat, c1aw, c1ab, styles, ic);
    demod_kernel<<<(4 * oc + 127) / 128, 128, 0, stream>>>(c1w, styles, demod, ic, oc);
    dim3 gc(4 * HW / 16, oc / 16);
    modconv3x3_kernel<<<gc, 32, 0, stream>>>(u, c1w, styles, demod, n1, tmp,
                                             ic, oc, H, H, (unsigned)(2 * i));
    // conv2: tmp -> u
    style_kernel<<<oc / 16, 32, 0, stream>>>(wlat, c2aw, c2ab, styles, oc);
    demod_kernel<<<(4 * oc + 127) / 128, 128, 0, stream>>>(c2w, styles, demod, oc, oc);
    modconv3x3_kernel<<<gc, 32, 0, stream>>>(tmp, c2w, styles, demod, n2, u,
                                             oc, oc, H, H, (unsigned)(2 * i + 1));
    xin = u;

    // toRGB + skip combine
    style_kernel<<<oc / 16, 32, 0, stream>>>(wlat, tgaw, tgab, styles, oc);
    torgb_kernel<<<(4 * HW + 255) / 256, 256, 0, stream>>>(xin, tgw, styles, rgbT, oc, HW);
    rgb_combine_kernel<<<(4 * HW + 255) / 256, 256, 0, stream>>>(rgb_cur, rgbT, rgb_alt,
                                                                 H, i == 0 ? 1 : 0);
    { float* t2 = rgb_cur; rgb_cur = rgb_alt; rgb_alt = t2; }

    if (i == 3) {  // self-attention at 64x64, ch=64
      sigma_kernel<<<1, 32, 0, stream>>>(
          (const float*)d_in[73], (const float*)d_in[75], (const float*)d_in[77],
          (const float*)d_in[79], (const float*)d_in[82], (const float*)d_in[83],
          (const float*)d_in[84], (const float*)d_in[85], sigma);
      qkv_kernel<<<(4 * 4096 + 255) / 256, 256, 0, stream>>>(
          xin, (const float*)d_in[73], (const float*)d_in[74],
          (const float*)d_in[75], (const float*)d_in[76],
          (const float*)d_in[77], (const float*)d_in[78],
          sigma, qbuf, kbuf, vbuf);
      dim3 ga(4096 / 16, 4);
      attn_flash_kernel<<<ga, 32, 0, stream>>>(qbuf, kbuf, vbuf, obuf);
      attn_out_kernel<<<(4 * 64 * 4096 + 255) / 256, 256, 0, stream>>>(
          xin, obuf, (const float*)d_in[79], (const float*)d_in[80],
          sigma, (const float*)d_in[81], tmp);
      xin = tmp;
    }
  }

  tanh_kernel<<<(4 * 128 * 128 + 255) / 256, 256, 0, stream>>>(rgb_cur, (float*)d_out,
                                                               4 * 128 * 128);
}